// ThreePhaseTerm_70128226009317
// MI455X (gfx1250) — compile-verified
//
#include <hip/hip_runtime.h>
#include <stdint.h>

// Three-phase astrochemistry RHS for MI455X (gfx1250).
// One workgroup (256 threads = 8 wave32) handles BPB=4 batch rows.
// y rows + dy rows live in LDS; scatters are ds_add_f32 atomics;
// row staging uses gfx1250 async global<->LDS transfers (ASYNCcnt);
// reaction constants are streamed as b128 (float4/int4) coalesced loads.

#define BLOCK 256
#define BPB   4
#define SMAX  1024
#define EPSV  1e-10f

__device__ __forceinline__ float wave_sum(float v) {
#pragma unroll
  for (int off = 16; off > 0; off >>= 1)
    v += __shfl_xor(v, off, 32);   // wave32 butterfly reduction
  return v;
}

__launch_bounds__(BLOCK)
__global__ void three_phase_rhs_kernel(
    const float* __restrict__ t_in, const float* __restrict__ y_in,
    const float* __restrict__ den_gas,
    const float* __restrict__ a1, const float* __restrict__ g1,
    const float* __restrict__ a2, const float* __restrict__ g2,
    const int* __restrict__ r1_1, const int* __restrict__ p_1,
    const int* __restrict__ r1_2, const int* __restrict__ r2_2,
    const int* __restrict__ p_2,
    const int* __restrict__ inds_surf, const int* __restrict__ inds_mant,
    float* __restrict__ out,
    int Bn, int Sn, int N1, int N2, int NS)
{
  __shared__ float y_lds[BPB][SMAX];   // 16 KB
  __shared__ float dy_lds[BPB][SMAX];  // 16 KB
  __shared__ float mask_f[SMAX];       // 4 KB  surface-membership mask (0/1)
  __shared__ float red[4 * BPB];       // gain[4], loss[4], ysurf[4], ymant[4]

  const int tid = threadIdx.x;
  const int b0  = blockIdx.x * BPB;

  // ---- zero dy rows, mask, reduction slots -------------------------------
  for (int i = tid; i < Sn; i += BLOCK) {
    mask_f[i] = 0.0f;
#pragma unroll
    for (int q = 0; q < BPB; ++q) dy_lds[q][i] = 0.0f;
  }
  if (tid < 4 * BPB) red[tid] = 0.0f;

  // ---- async copy y rows into LDS (16B per lane per row) -----------------
#pragma unroll
  for (int q = 0; q < BPB; ++q) {
    int b = b0 + q; if (b >= Bn) b = Bn - 1;
    for (int c = tid * 4; c < Sn; c += BLOCK * 4) {
      unsigned long long gaddr = (unsigned long long)(uintptr_t)(y_in + (size_t)b * Sn + c);
      unsigned lds_off = (unsigned)(uintptr_t)&y_lds[q][c];   // low 32 bits = LDS offset
      asm volatile("global_load_async_to_lds_b128 %0, %1, off"
                   :: "v"(lds_off), "v"(gaddr) : "memory");
    }
  }
  asm volatile("s_wait_asynccnt 0x0" ::: "memory");
  __syncthreads();

  // ---- surface membership mask (general: uses inds_surf values) ---------
  for (int i = tid; i < NS; i += BLOCK) mask_f[inds_surf[i]] = 1.0f;
  __syncthreads();

  // ---- per-row medium parameters (uniform scalar loads) ------------------
  float invT[BPB], sq[BPB], dg[BPB];
#pragma unroll
  for (int q = 0; q < BPB; ++q) {
    int b = b0 + q; if (b >= Bn) b = Bn - 1;
    float t   = t_in[b];
    float sig = 1.0f / (1.0f + __expf(-t));
    float T   = 10.0f + 5.0f * sig;
    invT[q] = 1.0f / T;
    sq[q]   = sqrtf(T * (1.0f / 300.0f));
    dg[q]   = den_gas[b];
  }

  float gacc[BPB] = {0.f, 0.f, 0.f, 0.f};
  float lacc[BPB] = {0.f, 0.f, 0.f, 0.f};

  // ---- first-order reactions: vectorized b128 constant streaming ---------
  {
    const int nvec = N1 >> 2;                       // groups of 4 reactions
    const float4* a4 = reinterpret_cast<const float4*>(a1);
    const float4* g4 = reinterpret_cast<const float4*>(g1);
    const int4*   r4 = reinterpret_cast<const int4*>(r1_1);
    const int4*   p4 = reinterpret_cast<const int4*>(p_1);
    for (int jv = tid; jv < nvec; jv += BLOCK) {
      float4 av = a4[jv], gv = g4[jv];
      int4   rv = r4[jv], pv = p4[jv];
      float  aj[4] = {av.x, av.y, av.z, av.w};
      float  gj[4] = {gv.x, gv.y, gv.z, gv.w};
      int    rj[4] = {rv.x, rv.y, rv.z, rv.w};
      int    pj[4] = {pv.x, pv.y, pv.z, pv.w};
#pragma unroll
      for (int k = 0; k < 4; ++k) {
        float mp = mask_f[pj[k]], mr = mask_f[rj[k]];
#pragma unroll
        for (int q = 0; q < BPB; ++q) {
          float term = aj[k] * __expf(-gj[k] * invT[q]) * y_lds[q][rj[k]];
          atomicAdd(&dy_lds[q][pj[k]],  term);
          atomicAdd(&dy_lds[q][rj[k]], -term);
          gacc[q] = fmaf(mp, term, gacc[q]);
          lacc[q] = fmaf(mr, term, lacc[q]);
        }
      }
    }
    for (int j = (nvec << 2) + tid; j < N1; j += BLOCK) {   // tail
      float a = a1[j], g = g1[j];
      int   r = r1_1[j], p = p_1[j];
      float mp = mask_f[p], mr = mask_f[r];
#pragma unroll
      for (int q = 0; q < BPB; ++q) {
        float term = a * __expf(-g * invT[q]) * y_lds[q][r];
        atomicAdd(&dy_lds[q][p],  term);
        atomicAdd(&dy_lds[q][r], -term);
        gacc[q] = fmaf(mp, term, gacc[q]);
        lacc[q] = fmaf(mr, term, lacc[q]);
      }
    }
  }

  // ---- second-order reactions: vectorized b128 constant streaming --------
  {
    const int nvec = N2 >> 2;
    const float4* a4 = reinterpret_cast<const float4*>(a2);
    const float4* g4 = reinterpret_cast<const float4*>(g2);
    const int4*   ra4 = reinterpret_cast<const int4*>(r1_2);
    const int4*   rb4 = reinterpret_cast<const int4*>(r2_2);
    const int4*   p4  = reinterpret_cast<const int4*>(p_2);
    for (int jv = tid; jv < nvec; jv += BLOCK) {
      float4 av = a4[jv], gv = g4[jv];
      int4   rav = ra4[jv], rbv = rb4[jv], pv = p4[jv];
      float  aj[4] = {av.x, av.y, av.z, av.w};
      float  gj[4] = {gv.x, gv.y, gv.z, gv.w};
      int    raj[4] = {rav.x, rav.y, rav.z, rav.w};
      int    rbj[4] = {rbv.x, rbv.y, rbv.z, rbv.w};
      int    pj[4]  = {pv.x, pv.y, pv.z, pv.w};
#pragma unroll
      for (int k = 0; k < 4; ++k) {
        float mp = mask_f[pj[k]], mr = mask_f[raj[k]] + mask_f[rbj[k]];
#pragma unroll
        for (int q = 0; q < BPB; ++q) {
          float rate = aj[k] * sq[q] * __expf(-gj[k] * invT[q]);
          float term = rate * y_lds[q][raj[k]] * y_lds[q][rbj[k]] * dg[q];
          atomicAdd(&dy_lds[q][pj[k]],   term);
          atomicAdd(&dy_lds[q][raj[k]], -term);
          atomicAdd(&dy_lds[q][rbj[k]], -term);
          gacc[q] = fmaf(mp, term, gacc[q]);
          lacc[q] = fmaf(mr, term, lacc[q]);
        }
      }
    }
    for (int j = (nvec << 2) + tid; j < N2; j += BLOCK) {   // tail
      float a = a2[j], g = g2[j];
      int   ra = r1_2[j], rb = r2_2[j], p = p_2[j];
      float mp = mask_f[p], mr = mask_f[ra] + mask_f[rb];
#pragma unroll
      for (int q = 0; q < BPB; ++q) {
        float rate = a * sq[q] * __expf(-g * invT[q]);
        float term = rate * y_lds[q][ra] * y_lds[q][rb] * dg[q];
        atomicAdd(&dy_lds[q][p],   term);
        atomicAdd(&dy_lds[q][ra], -term);
        atomicAdd(&dy_lds[q][rb], -term);
        gacc[q] = fmaf(mp, term, gacc[q]);
        lacc[q] = fmaf(mr, term, lacc[q]);
      }
    }
  }

  // ---- surface / mantle abundance totals --------------------------------
  float ys[BPB] = {0.f, 0.f, 0.f, 0.f};
  float ym[BPB] = {0.f, 0.f, 0.f, 0.f};
  for (int i = tid; i < NS; i += BLOCK) {
    int si = inds_surf[i], mi = inds_mant[i];
#pragma unroll
    for (int q = 0; q < BPB; ++q) { ys[q] += y_lds[q][si]; ym[q] += y_lds[q][mi]; }
  }

  // ---- block reduction of 16 scalars (wave32 shuffles + 8 LDS atomics) ---
  {
    float v[4 * BPB];
#pragma unroll
    for (int q = 0; q < BPB; ++q) {
      v[q]           = gacc[q];
      v[BPB + q]     = lacc[q];
      v[2 * BPB + q] = ys[q];
      v[3 * BPB + q] = ym[q];
    }
#pragma unroll
    for (int k = 0; k < 4 * BPB; ++k) {
      float s = wave_sum(v[k]);
      if ((tid & 31) == 0) atomicAdd(&red[k], s);
    }
  }
  __syncthreads();

  // ---- transfer coefficients --------------------------------------------
  float ks[BPB], km[BPB];
#pragma unroll
  for (int q = 0; q < BPB; ++q) {
    ks[q] = red[q]       / (red[2 * BPB + q] + EPSV);  // k_s2m = gain / (ysurf+eps)
    km[q] = red[BPB + q] / (red[3 * BPB + q] + EPSV);  // k_m2s = loss / (ymant+eps)
  }

  // ---- surface <-> mantle transfer (atomic: tolerate repeated indices) ---
  for (int i = tid; i < NS; i += BLOCK) {
    int si = inds_surf[i], mi = inds_mant[i];
#pragma unroll
    for (int q = 0; q < BPB; ++q) {
      float s2m = ks[q] * y_lds[q][si];
      float m2s = km[q] * y_lds[q][mi];
      atomicAdd(&dy_lds[q][si], m2s - s2m);
      atomicAdd(&dy_lds[q][mi], s2m - m2s);
    }
  }
  __syncthreads();

  // ---- async store dy rows LDS -> global ---------------------------------
#pragma unroll
  for (int q = 0; q < BPB; ++q) {
    int b = b0 + q;
    if (b < Bn) {
      for (int c = tid * 4; c < Sn; c += BLOCK * 4) {
        unsigned long long gaddr = (unsigned long long)(uintptr_t)(out + (size_t)b * Sn + c);
        unsigned lds_off = (unsigned)(uintptr_t)&dy_lds[q][c];
        asm volatile("global_store_async_from_lds_b128 %0, %1, off"
                     :: "v"(gaddr), "v"(lds_off) : "memory");
      }
    }
  }
  asm volatile("s_wait_asynccnt 0x0" ::: "memory");
}

extern "C" void kernel_launch(void* const* d_in, const int* in_sizes, int n_in,
                              void* d_out, int out_size, void* d_ws, size_t ws_size,
                              hipStream_t stream) {
  const float* t_in      = (const float*)d_in[0];
  const float* y_in      = (const float*)d_in[1];
  const float* den_gas   = (const float*)d_in[2];
  const float* alpha_1   = (const float*)d_in[3];
  const float* gamma_1   = (const float*)d_in[4];
  const float* alpha_2   = (const float*)d_in[5];
  const float* gamma_2   = (const float*)d_in[6];
  const int*   r1_1st    = (const int*)d_in[7];
  const int*   p_1st     = (const int*)d_in[8];
  const int*   r1_2nd    = (const int*)d_in[9];
  const int*   r2_2nd    = (const int*)d_in[10];
  const int*   p_2nd     = (const int*)d_in[11];
  const int*   inds_surf = (const int*)d_in[12];
  const int*   inds_mant = (const int*)d_in[13];
  float*       out       = (float*)d_out;

  const int Bn = in_sizes[0];
  const int Sn = in_sizes[1] / (Bn > 0 ? Bn : 1);
  const int N1 = in_sizes[3];
  const int N2 = in_sizes[5];
  const int NS = in_sizes[12];

  const int grid = (Bn + BPB - 1) / BPB;
  three_phase_rhs_kernel<<<dim3(grid), dim3(BLOCK), 0, stream>>>(
      t_in, y_in, den_gas, alpha_1, gamma_1, alpha_2, gamma_2,
      r1_1st, p_1st, r1_2nd, r2_2nd, p_2nd, inds_surf, inds_mant,
      out, Bn, Sn, N1, N2, NS);
}